// VMD_58110907515012
// MI455X (gfx1250) — compile-verified
//
#include <hip/hip_runtime.h>

#define T_LEN   1048576
#define KMODES  3
#define ALPHA_C 2000.0f
#define TAU_C   1e-7f
#define TOL_C   1e-6f
#define NITER   50
#define TWO_PI  6.283185307179586476925286766559f

typedef float v2f __attribute__((ext_vector_type(2)));
typedef float v8f __attribute__((ext_vector_type(8)));

// Interleaved {cos, sin}(2*pi*k/16), k = 0..15 — the radix-16 DFT matrix
// entries W16^{jr} = cos - i*sin indexed by (j*r) mod 16. One cacheline.
__constant__ float W16TAB[32] = {
   1.0f,                 0.0f,
   0.92387953251128675f, 0.38268343236508977f,
   0.70710678118654752f, 0.70710678118654752f,
   0.38268343236508977f, 0.92387953251128675f,
   0.0f,                 1.0f,
  -0.38268343236508977f, 0.92387953251128675f,
  -0.70710678118654752f, 0.70710678118654752f,
  -0.92387953251128675f, 0.38268343236508977f,
  -1.0f,                 0.0f,
  -0.92387953251128675f,-0.38268343236508977f,
  -0.70710678118654752f,-0.70710678118654752f,
  -0.38268343236508977f,-0.92387953251128675f,
   0.0f,                -1.0f,
   0.38268343236508977f,-0.92387953251128675f,
   0.70710678118654752f,-0.70710678118654752f,
   0.92387953251128675f,-0.38268343236508977f,
};

// ---------------------------------------------------------------------------
// Radix-16 Stockham DIF stage computed with V_WMMA_F32_16X16X4_F32.
//   y[q + s*(16*p + j)] = W_n^{j*p} * sum_r x[q + s*(p + m*r)] * W16^{j*r}
// Each wave handles 16 (p,q) pairs as the 16 columns of a 16x16 complex GEMM:
//   Y = W16 * X,  complex via 4 real K=16 matmuls = 16 wmma_f32_16x16x4 ops.
// blockIdx.y = batch (mode) with stride T_LEN complex elements.
// ---------------------------------------------------------------------------
__global__ __launch_bounds__(256)
void fft16_stage(const float2* __restrict__ src, float2* __restrict__ dst,
                 int n, int sShift, int m)
{
  const int s    = 1 << sShift;
  const int lane = threadIdx.x & 31;
  const int hi   = lane >> 4;              // half-wave select
  const int lo   = lane & 15;
  const int wave = blockIdx.x * (blockDim.x >> 5) + (threadIdx.x >> 5);
  const int colBase = wave << 4;           // 16 (p,q) pairs per wave

  const float2* sb = src + (size_t)blockIdx.y * (size_t)T_LEN;
  float2*       db = dst + (size_t)blockIdx.y * (size_t)T_LEN;

  // ---- B operands: X, gathered per ISA B-layout (VGPR v: K=v / K=v+2) ----
  v2f Br[4], Bi[4];
  {
    const int g = colBase + lo;            // this lane's column pair id
    const int p = g >> sShift;
    const int q = g & (s - 1);
#pragma unroll
    for (int kc = 0; kc < 4; ++kc) {
#pragma unroll
      for (int v = 0; v < 2; ++v) {
        const int r = 4 * kc + 2 * hi + v; // global K row 0..15
        const size_t idx = (size_t)q + (size_t)s * ((size_t)p + (size_t)m * (size_t)r);
        const float2 z = sb[idx];          // global_load_b64
        Br[kc][v] = z.x;
        Bi[kc][v] = z.y;
      }
    }
  }

  // ---- A operands: W16[j][r] from constant table (no runtime sincos) ----
  v2f Ar[4], Ai[4], AiN[4];
  {
    const int j = lo;                      // A: M = lane & 15
#pragma unroll
    for (int kc = 0; kc < 4; ++kc) {
#pragma unroll
      for (int v = 0; v < 2; ++v) {
        const int r  = 4 * kc + 2 * hi + v;
        const int e  = 2 * ((j * r) & 15);
        const float cs = W16TAB[e];
        const float sn = W16TAB[e + 1];
        Ar[kc][v]  = cs;    // Re(W16)
        Ai[kc][v]  = -sn;   // Im(W16)
        AiN[kc][v] = sn;    // -Im(W16)  (F32 WMMA has no A/B NEG modifier)
      }
    }
  }

  // ---- complex 16x16x16 matmul as 16 chained wmma (K accumulated 4 at a
  //      time); B operands used in consecutive instructions ----
  v8f yr = {0.f,0.f,0.f,0.f,0.f,0.f,0.f,0.f};
  v8f yi = {0.f,0.f,0.f,0.f,0.f,0.f,0.f,0.f};
#pragma unroll
  for (int kc = 0; kc < 4; ++kc) {
    yr = __builtin_amdgcn_wmma_f32_16x16x4_f32(false, Ar[kc],  false, Br[kc], (short)0, yr, false, false);
    yi = __builtin_amdgcn_wmma_f32_16x16x4_f32(false, Ai[kc],  false, Br[kc], (short)0, yi, false, false);
    yr = __builtin_amdgcn_wmma_f32_16x16x4_f32(false, AiN[kc], false, Bi[kc], (short)0, yr, false, false);
    yi = __builtin_amdgcn_wmma_f32_16x16x4_f32(false, Ar[kc],  false, Bi[kc], (short)0, yi, false, false);
  }

  // ---- stage twiddle W_n^{j*p} + scatter (D layout: M = v + 8*hi, N = lo) ----
  {
    const int g = colBase + lo;
    const int p = g >> sShift;
    const int q = g & (s - 1);
    const float invN = 1.0f / (float)n;
#pragma unroll
    for (int v = 0; v < 8; ++v) {
      const int j  = v + 8 * hi;
      const int jp = (j * p) & (n - 1);    // n is a power of two
      const float ph = -TWO_PI * (float)jp * invN;
      const float tr = __cosf(ph);
      const float ti = __sinf(ph);
      float2 o;
      o.x = yr[v] * tr - yi[v] * ti;
      o.y = yr[v] * ti + yi[v] * tr;
      const size_t di = (size_t)q + (size_t)s * ((size_t)(p << 4) + (size_t)j);
      db[di] = o;                          // global_store_b64
    }
  }
}

// ---------------------------------------------------------------------------
__global__ __launch_bounds__(256)
void prep_input(const float* __restrict__ x, float2* __restrict__ cplx)
{
  const int i = blockIdx.x * blockDim.x + threadIdx.x;
  float2 z; z.x = x[i]; z.y = 0.0f;
  cplx[i] = z;
}

__global__ __launch_bounds__(256)
void zero_f32(float* __restrict__ p, int n)
{
  int i = blockIdx.x * blockDim.x + threadIdx.x;
  const int stride = gridDim.x * blockDim.x;
  for (; i < n; i += stride) p[i] = 0.0f;
}

__global__ void init_state(const float* __restrict__ omega_init,
                           float* __restrict__ omega, int* __restrict__ done)
{
  if (threadIdx.x == 0) {
    omega[0] = omega_init[0];
    omega[1] = omega_init[1];
    omega[2] = omega_init[2];
    *done = 0;
  }
}

// fftshift (== ifftshift for even T): out[i] = in[(i + T/2) mod T]
__global__ __launch_bounds__(256)
void fftshift_c(const float2* __restrict__ in, float2* __restrict__ out)
{
  const int i = blockIdx.x * blockDim.x + threadIdx.x;
  const int j = (i + T_LEN / 2) & (T_LEN - 1);
  out[i] = in[j];
}

// work[k][i] = conj(ifftshift(uhat[k])[i])  (ifft via conj-fft-conj)
__global__ __launch_bounds__(256)
void pre_ifft(const float2* __restrict__ uhat, float2* __restrict__ work)
{
  const int i = blockIdx.x * blockDim.x + threadIdx.x;
  const float2* u = uhat + (size_t)blockIdx.y * (size_t)T_LEN;
  float2*       w = work + (size_t)blockIdx.y * (size_t)T_LEN;
  const int j = (i + T_LEN / 2) & (T_LEN - 1);
  float2 z = u[j];
  z.y = -z.y;
  w[i] = z;
}

// imf[k][i] = Re(conj(fftres)/T) = Re(fftres)/T
__global__ __launch_bounds__(256)
void post_ifft(const float2* __restrict__ work, float* __restrict__ out)
{
  const int i = blockIdx.x * blockDim.x + threadIdx.x;
  const float2* w = work + (size_t)blockIdx.y * (size_t)T_LEN;
  out[(size_t)blockIdx.y * (size_t)T_LEN + i] = w[i].x * (1.0f / (float)T_LEN);
}

// ---------------------------------------------------------------------------
// One VMD fixed-point iteration (elementwise part + block-level partial sums).
// partials[block][0..7] = {sum|du|^2, sum|u_prev|^2, pw0,pw1,pw2, fpw0,fpw1,fpw2}
// ---------------------------------------------------------------------------
__global__ __launch_bounds__(256)
void vmd_update(const float2* __restrict__ fhat, float2* __restrict__ uhat,
                float2* __restrict__ lam, const float* __restrict__ omega,
                const int* __restrict__ done, float* __restrict__ partials)
{
  if (*done) return;                        // frozen: keep state (matches where(done,...))
  const int t = blockIdx.x * blockDim.x + threadIdx.x;
  const float f = (float)t * (1.0f / (float)T_LEN) - 0.5f;

  const float w0 = omega[0], w1 = omega[1], w2 = omega[2];
  const float b0 = (f - w0) * (f - w0);
  const float b1 = (f - w1) * (f - w1);
  const float b2 = (f - w2) * (f - w2);
  const float tt = TAU_C * TAU_C;
  const float d0 = b0 + b1 + tt;            // mode 0: + right neighbor
  const float d1 = b1 + b0 + b2 + tt;       // mode 1: + both neighbors
  const float d2 = b2 + b1 + tt;            // mode 2: + left neighbor

  const float2 fz = fhat[t];
  const float2 lz = lam[t];
  const float nr = fz.x - 0.5f * lz.x;
  const float ni = fz.y - 0.5f * lz.y;

  const float i0 = 1.0f / (1.0f + ALPHA_C * d0);
  const float i1 = 1.0f / (1.0f + ALPHA_C * d1);
  const float i2 = 1.0f / (1.0f + ALPHA_C * d2);

  const float u0r = nr * i0, u0i = ni * i0;
  const float u1r = nr * i1, u1i = ni * i1;
  const float u2r = nr * i2, u2i = ni * i2;

  float2* p0 = uhat + (size_t)t;
  float2* p1 = p0 + (size_t)T_LEN;
  float2* p2 = p1 + (size_t)T_LEN;
  const float2 o0 = *p0;
  const float2 o1 = *p1;
  const float2 o2 = *p2;

  const float snum = (u0r-o0.x)*(u0r-o0.x) + (u0i-o0.y)*(u0i-o0.y)
                   + (u1r-o1.x)*(u1r-o1.x) + (u1i-o1.y)*(u1i-o1.y)
                   + (u2r-o2.x)*(u2r-o2.x) + (u2i-o2.y)*(u2i-o2.y);
  const float sden = o0.x*o0.x + o0.y*o0.y + o1.x*o1.x + o1.y*o1.y
                   + o2.x*o2.x + o2.y*o2.y;

  const float pw0 = u0r*u0r + u0i*u0i;
  const float pw1 = u1r*u1r + u1i*u1i;
  const float pw2 = u2r*u2r + u2i*u2i;

  float2 n0; n0.x = u0r; n0.y = u0i; *p0 = n0;
  float2 n1; n1.x = u1r; n1.y = u1i; *p1 = n1;
  float2 n2; n2.x = u2r; n2.y = u2i; *p2 = n2;

  float2 lnew;
  lnew.x = lz.x + TAU_C * ((u0r + u1r + u2r) - fz.x);
  lnew.y = lz.y + TAU_C * ((u0i + u1i + u2i) - fz.y);
  lam[t] = lnew;

  __shared__ float red[256 * 8];
  float vals[8] = {snum, sden, pw0, pw1, pw2, f * pw0, f * pw1, f * pw2};
#pragma unroll
  for (int i = 0; i < 8; ++i) red[threadIdx.x * 8 + i] = vals[i];
  __syncthreads();
  for (int st = 128; st > 0; st >>= 1) {
    if ((int)threadIdx.x < st) {
#pragma unroll
      for (int i = 0; i < 8; ++i)
        red[threadIdx.x * 8 + i] += red[(threadIdx.x + st) * 8 + i];
    }
    __syncthreads();
  }
  if (threadIdx.x == 0) {
#pragma unroll
    for (int i = 0; i < 8; ++i) partials[blockIdx.x * 8 + i] = red[i];
  }
}

// Finish reductions, update omega, evaluate convergence freeze flag.
__global__ __launch_bounds__(256)
void vmd_reduce(float* __restrict__ omega, int* __restrict__ done,
                const float* __restrict__ partials, int nblocks, int iter)
{
  if (*done) return;
  float acc[8] = {0.f,0.f,0.f,0.f,0.f,0.f,0.f,0.f};
  for (int b = threadIdx.x; b < nblocks; b += blockDim.x) {
#pragma unroll
    for (int i = 0; i < 8; ++i) acc[i] += partials[b * 8 + i];
  }
  __shared__ float red[256 * 8];
#pragma unroll
  for (int i = 0; i < 8; ++i) red[threadIdx.x * 8 + i] = acc[i];
  __syncthreads();
  for (int st = 128; st > 0; st >>= 1) {
    if ((int)threadIdx.x < st) {
#pragma unroll
      for (int i = 0; i < 8; ++i)
        red[threadIdx.x * 8 + i] += red[(threadIdx.x + st) * 8 + i];
    }
    __syncthreads();
  }
  if (threadIdx.x == 0) {
    const float snum = red[0], sden = red[1];
    const float o0 = red[5] / red[2];
    const float o1 = red[6] / red[3];
    const float o2 = red[7] / red[4];
    // om_diff = mean_k |omega_new[k] - roll(omega_new,1)[k]|
    const float omd = (fabsf(o0 - o2) + fabsf(o1 - o0) + fabsf(o2 - o1)) * (1.0f / 3.0f);
    const float ud  = snum / sden;
    omega[0] = o0; omega[1] = o1; omega[2] = o2;
    if ((iter % 10 == 0) && (ud < TOL_C) && (omd < TOL_C)) *done = 1;
  }
}

// ---------------------------------------------------------------------------
extern "C" void kernel_launch(void* const* d_in, const int* in_sizes, int n_in,
                              void* d_out, int out_size, void* d_ws, size_t ws_size,
                              hipStream_t stream)
{
  (void)in_sizes; (void)n_in; (void)out_size; (void)ws_size;
  const float* x          = (const float*)d_in[0];
  const float* omega_init = (const float*)d_in[1];
  float* out = (float*)d_out;

  // workspace carve-up (floats); total ~92 MB — fits L2-resident (192 MB)
  float* ws = (float*)d_ws;
  size_t off = 0;
  float2* bigA = (float2*)(ws + off); off += (size_t)2 * KMODES * T_LEN;  // FFT ping
  float2* bigB = (float2*)(ws + off); off += (size_t)2 * KMODES * T_LEN;  // FFT pong
  float2* fhat = (float2*)(ws + off); off += (size_t)2 * T_LEN;           // fftshift(fft(x))
  float2* uhat = (float2*)(ws + off); off += (size_t)2 * KMODES * T_LEN;  // modes (complex)
  float2* lam  = (float2*)(ws + off); off += (size_t)2 * T_LEN;           // dual (complex)
  float* partials = ws + off; off += (size_t)4096 * 8;
  float* omega    = ws + off; off += 8;
  int*   done     = (int*)(ws + off);

  const int TB = 256;
  const int gT = T_LEN / TB;                 // 4096 blocks, exact cover

  prep_input<<<gT, TB, 0, stream>>>(x, bigA);
  zero_f32<<<1024, TB, 0, stream>>>((float*)uhat, 2 * KMODES * T_LEN);
  zero_f32<<<1024, TB, 0, stream>>>((float*)lam, 2 * T_LEN);
  init_state<<<1, 32, 0, stream>>>(omega_init, omega, done);

  // forward FFT: 5 radix-16 Stockham stages (WMMA f32 16x16x4)
  {
    float2* s0 = bigA; float2* s1 = bigB;
    int n = T_LEN;
    for (int st = 0; st < 5; ++st) {
      fft16_stage<<<dim3(512, 1), TB, 0, stream>>>(s0, s1, n, 4 * st, n >> 4);
      float2* tmp = s0; s0 = s1; s1 = tmp;
      n >>= 4;
    }
    fftshift_c<<<gT, TB, 0, stream>>>(s0 /* == bigB */, fhat);
  }

  // 50 fixed-point iterations with freeze-on-convergence
  for (int it = 0; it < NITER; ++it) {
    vmd_update<<<gT, TB, 0, stream>>>(fhat, uhat, lam, omega, done, partials);
    vmd_reduce<<<1, TB, 0, stream>>>(omega, done, partials, gT, it);
  }

  // inverse FFT of all K modes, batched via blockIdx.y
  pre_ifft<<<dim3(gT, KMODES), TB, 0, stream>>>(uhat, bigA);
  {
    float2* s0 = bigA; float2* s1 = bigB;
    int n = T_LEN;
    for (int st = 0; st < 5; ++st) {
      fft16_stage<<<dim3(512, KMODES), TB, 0, stream>>>(s0, s1, n, 4 * st, n >> 4);
      float2* tmp = s0; s0 = s1; s1 = tmp;
      n >>= 4;
    }
    post_ifft<<<dim3(gT, KMODES), TB, 0, stream>>>(s0 /* == bigB */, out);
  }
}